// PPO_46291157516504
// MI455X (gfx1250) — compile-verified
//
#include <hip/hip_runtime.h>
#include <hip/hip_bf16.h>

#define GAMMA    0.99f
#define LMBDA    0.95f
#define EPS_CLIP 0.2f
#define VF_COEF  0.5f
#define ENT_COEF 0.01f
#define KL_COEF  0.01f
#define EPSV     1e-10f

// Problem shape fixed by the reference.
#define BB 4096
#define TT 512
#define AA 16

typedef __attribute__((ext_vector_type(2))) float v2f;
typedef __attribute__((ext_vector_type(8))) float v8f;

// ---------------------------------------------------------------------------
// WMMA-based 4-way reduction (CDNA5 V_WMMA_F32_16X16X4_F32, f32 in / f32 out).
//
// src holds `count` partial-sets of 4 floats: src[i*4 + q], q in 0..3.
// With B = all-ones, D[m][n] = sum_k A[m][k] + C[m][n], so the matrix unit is
// a 64-wide f32 adder tree. Quantity q is fed only into rows {q,q+4,q+8,q+12}
// (row = lane%16, quantity = row%4), so the four sums stay separate in the
// 16x16 accumulator. Extraction: c[q]+c[q+4] covers rows q,q+4 on lanes 0-15
// and rows q+8,q+12 on lanes 16-31; one shfl_xor(16) combines the halves.
// Must be called by one full wave (EXEC all ones). Result broadcast to lanes.
// ---------------------------------------------------------------------------
__device__ __forceinline__ void wmma_reduce4(const float* __restrict__ src,
                                             int count, int lane, float out[4]) {
  const int r  = lane & 15;            // row in A (16x4)
  const int q  = r & 3;                // which quantity this row carries
  const int s  = r >> 2;               // sub-stream 0..3
  const int k0 = (lane < 16) ? 0 : 2;  // A layout: lanes 0-15 hold K=0,1; 16-31 hold K=2,3
  v2f b; b.x = 1.0f; b.y = 1.0f;       // B = ones (4x16); layout irrelevant, all elems equal
  v8f c = {};                          // C/D accumulator, 16x16 f32
  const int iters = (count + 15) >> 4; // each WMMA consumes 16 partial-sets
  for (int it = 0; it < iters; ++it) {
    const int i0 = it * 16 + s * 4 + k0;
    const int i1 = i0 + 1;
    v2f a;
    a.x = (i0 < count) ? src[i0 * 4 + q] : 0.0f;
    a.y = (i1 < count) ? src[i1 * 4 + q] : 0.0f;
    // 8-arg form: (neg_a, A, neg_b, B, c_mod, C, reuse_a, reuse_b)
    c = __builtin_amdgcn_wmma_f32_16x16x4_f32(false, a, false, b,
                                              (short)0, c, false, false);
  }
#pragma unroll
  for (int qq = 0; qq < 4; ++qq) {
    float t = c[qq] + c[qq + 4];       // rows qq,qq+4 (lo lanes) / qq+8,qq+12 (hi lanes)
    t += __shfl_xor(t, 16, 32);        // combine lane halves -> total in every lane
    out[qq] = t;
  }
}

// ---------------------------------------------------------------------------
// Kernel 1: reversed-time GAE scan, one thread per batch row.
// Writes raw advantages; accumulates masked sum(adv), sum(adv^2), sum(valid).
// ---------------------------------------------------------------------------
__global__ void ppo_gae_kernel(const float* __restrict__ values,
                               const float* __restrict__ vprime,
                               const float* __restrict__ r,
                               const unsigned char* __restrict__ done,
                               float* __restrict__ adv,
                               float* __restrict__ k1_partials) {
  const int b = blockIdx.x * blockDim.x + threadIdx.x;
  float sa = 0.0f, sa2 = 0.0f, sv = 0.0f;
  if (b < BB) {
    const size_t base = (size_t)b * TT;
    float g = 0.0f;
    for (int t = TT - 1; t >= 0; --t) {
      const size_t e = base + t;
      const float nd    = done[e] ? 0.0f : 1.0f;
      const float td    = fmaf(GAMMA * nd, vprime[e], r[e]);
      const float delta = td - values[e];
      g = fmaf((GAMMA * LMBDA) * nd, g, delta);
      adv[e] = g;
      const float valid = (t == 0) ? 1.0f : (done[e - 1] ? 0.0f : 1.0f);
      sa  += g * valid;
      sa2 += g * g * valid;
      sv  += valid;
    }
  }
  __shared__ float lds[256 * 4];
  const int tid = threadIdx.x;
  lds[tid * 4 + 0] = sa;
  lds[tid * 4 + 1] = sa2;
  lds[tid * 4 + 2] = sv;
  lds[tid * 4 + 3] = 0.0f;
  __syncthreads();
  if (tid < 32) {
    float tot[4];
    wmma_reduce4(lds, 256, tid, tot);
    if (tid == 0) {
      float* o = k1_partials + (size_t)blockIdx.x * 4;
      o[0] = tot[0]; o[1] = tot[1]; o[2] = tot[2]; o[3] = 0.0f;
    }
  }
}

// ---------------------------------------------------------------------------
// Kernel 2: finalize masked mean / inv-std / vsum (single thread; 16 partials).
// var = E[a^2] - mean^2 over the masked set (algebraically same as reference).
// ---------------------------------------------------------------------------
__global__ void ppo_stats_kernel(const float* __restrict__ k1_partials,
                                 int nblocks, float* __restrict__ stats) {
  if (blockIdx.x == 0 && threadIdx.x == 0) {
    float sa = 0.0f, sa2 = 0.0f, sv = 0.0f;
    for (int i = 0; i < nblocks; ++i) {
      sa  += k1_partials[i * 4 + 0];
      sa2 += k1_partials[i * 4 + 1];
      sv  += k1_partials[i * 4 + 2];
    }
    const float mean = sa / sv;
    float var = sa2 / sv - mean * mean;
    var = fmaxf(var, 0.0f);
    stats[0] = mean;
    stats[1] = 1.0f / (sqrtf(var) + 1e-8f);
    stats[2] = sv;
  }
}

// ---------------------------------------------------------------------------
// Kernel 3: main elementwise pass over B*T elements (grid-stride).
// Accumulates {policy, value, entropy(p*log p), kl} and block-reduces them
// through LDS + the WMMA adder-tree.
// ---------------------------------------------------------------------------
__global__ void ppo_main_kernel(const float* __restrict__ pi,
                                const float* __restrict__ values,
                                const float* __restrict__ vprime,
                                const float* __restrict__ r,
                                const float* __restrict__ prob_a,
                                const int* __restrict__ a,
                                const unsigned char* __restrict__ done,
                                const float* __restrict__ adv,
                                const float* __restrict__ stats,
                                float* __restrict__ k3_partials) {
  const float mean    = stats[0];
  const float inv_std = stats[1];
  const int   N       = BB * TT;
  const int   stride  = gridDim.x * blockDim.x;

  float pol = 0.0f, vl = 0.0f, ent = 0.0f, kl = 0.0f;

  for (int e = blockIdx.x * blockDim.x + threadIdx.x; e < N; e += stride) {
    const int t = e & (TT - 1);                       // T is a power of two
    const unsigned char dm = done[e];
    const float valid = (t == 0) ? 1.0f : (done[e - 1] ? 0.0f : 1.0f);
    const float nd    = dm ? 0.0f : 1.0f;

    // Huber value loss: diff = values - td_target = -delta
    const float td   = fmaf(GAMMA * nd, vprime[e], r[e]);
    const float diff = values[e] - td;
    const float ad   = fabsf(diff);
    const float sl   = (ad < 1.0f) ? 0.5f * diff * diff : ad - 0.5f;
    vl += sl * valid;

    // clipped policy surrogate + approx KL
    const float advn = (adv[e] - mean) * inv_std;
    const int   ai   = a[e];
    const float* pe  = pi + (size_t)e * AA;
    const float pia  = pe[ai];
    const float lp   = __logf(pia + EPSV);
    const float lpo  = __logf(prob_a[e] + EPSV);
    const float d    = lp - lpo;
    const float ratio = __expf(d);
    kl += pia * d * valid;
    const float s1 = ratio * advn;
    const float rc = fminf(fmaxf(ratio, 1.0f - EPS_CLIP), 1.0f + EPS_CLIP);
    const float s2 = rc * advn;
    pol -= fminf(s1, s2) * valid;

    // entropy term: accumulate sum_a p*log(p)  (entropy_loss = mean of this)
    const float4* p4 = reinterpret_cast<const float4*>(pe);
    float plg = 0.0f;
#pragma unroll
    for (int j = 0; j < AA / 4; ++j) {
      const float4 p = p4[j];
      plg += (p.x > 0.0f) ? p.x * __logf(p.x) : 0.0f;
      plg += (p.y > 0.0f) ? p.y * __logf(p.y) : 0.0f;
      plg += (p.z > 0.0f) ? p.z * __logf(p.z) : 0.0f;
      plg += (p.w > 0.0f) ? p.w * __logf(p.w) : 0.0f;
    }
    ent += plg;
  }

  __shared__ float lds[256 * 4];
  const int tid = threadIdx.x;
  lds[tid * 4 + 0] = pol;
  lds[tid * 4 + 1] = vl;
  lds[tid * 4 + 2] = ent;
  lds[tid * 4 + 3] = kl;
  __syncthreads();
  if (tid < 32) {
    float tot[4];
    wmma_reduce4(lds, 256, tid, tot);
    if (tid == 0) {
      float* o = k3_partials + (size_t)blockIdx.x * 4;
      o[0] = tot[0]; o[1] = tot[1]; o[2] = tot[2]; o[3] = tot[3];
    }
  }
}

// ---------------------------------------------------------------------------
// Kernel 4: final WMMA-chain reduction of block partials + scalar combine.
// Launched with exactly one wave (32 threads).
// ---------------------------------------------------------------------------
__global__ void ppo_final_kernel(const float* __restrict__ k3_partials, int nb,
                                 const float* __restrict__ stats,
                                 float* __restrict__ out) {
  float tot[4];
  wmma_reduce4(k3_partials, nb, threadIdx.x, tot);
  if (threadIdx.x == 0) {
    const float vsum    = stats[2];
    const float invN    = 1.0f / (float)(BB * TT);
    const float policy  = tot[0] / vsum;
    const float value   = tot[1] / vsum;
    const float entloss = tot[2] * invN;    // -entropy.mean() == mean(p*log p)
    const float akl     = tot[3] / vsum;
    out[0] = policy + VF_COEF * value + ENT_COEF * entloss + KL_COEF * akl;
  }
}

// ---------------------------------------------------------------------------
extern "C" void kernel_launch(void* const* d_in, const int* in_sizes, int n_in,
                              void* d_out, int out_size, void* d_ws, size_t ws_size,
                              hipStream_t stream) {
  // setup_inputs() order: pi, values, values_prime, r, prob_a, a, done_mask
  const float*         pi     = (const float*)d_in[0];
  const float*         values = (const float*)d_in[1];
  const float*         vprime = (const float*)d_in[2];
  const float*         r      = (const float*)d_in[3];
  const float*         prob_a = (const float*)d_in[4];
  const int*           a      = (const int*)d_in[5];
  const unsigned char* done   = (const unsigned char*)d_in[6]; // jax bool = 1 byte

  const int BT = BB * TT;

  // Workspace layout (floats): adv[BT] | k1 partials[16*4] | stats[16] | k3 partials[2048*4]
  float* ws       = (float*)d_ws;
  float* adv      = ws;
  float* k1p      = ws + BT;
  float* stats    = k1p + 16 * 4;
  float* k3p      = stats + 16;

  const int k1_blocks = BB / 256;   // 16
  const int k3_blocks = 2048;

  ppo_gae_kernel<<<k1_blocks, 256, 0, stream>>>(values, vprime, r, done, adv, k1p);
  ppo_stats_kernel<<<1, 1, 0, stream>>>(k1p, k1_blocks, stats);
  ppo_main_kernel<<<k3_blocks, 256, 0, stream>>>(pi, values, vprime, r, prob_a,
                                                 a, done, adv, stats, k3p);
  ppo_final_kernel<<<1, 32, 0, stream>>>(k3p, k3_blocks, stats, (float*)d_out);
}